// Gaussian_Fit_78383153152165
// MI455X (gfx1250) — compile-verified
//
#include <hip/hip_runtime.h>
#include <hip/hip_bf16.h>

typedef float v2f __attribute__((ext_vector_type(2)));
typedef float v8f __attribute__((ext_vector_type(8)));

#define NN 64
#define HW 4096
#define XS 68    // padded row stride for 64-wide tiles (keeps b64 loads 8B-aligned, de-phases banks)
#define MS 132   // padded row stride for the 64x128 augmented [cov | I] matrix

__global__ __launch_bounds__(256) void gaussian_fit_kernel(
    const float* __restrict__ x, const float* __restrict__ eps,
    float* __restrict__ out)
{
  __shared__ float sXc[NN * XS];   // centered map Xc
  __shared__ float sTn[NN * XS];   // row-centered Tn; later reused as T1 = Xc @ Inv
  __shared__ float sM [NN * MS];   // augmented [cov | I] -> [I | inv]
  __shared__ float sRed[256];
  __shared__ float sRowM[NN];      // row means, later reused as pivot-column cache
  __shared__ float sPr;

  const int tid  = threadIdx.x;
  const int lane = tid & 31;
  const int wave = tid >> 5;
  const int l16  = lane & 15;      // M/N within tile
  const int lhi  = lane >> 4;      // K-pair / row-half selector

  const size_t gbase = (size_t)blockIdx.x * HW;

  // ---------------- Phase A: normalize, center, row-center ----------------
  float r[16];
  float ls = 0.f;
  #pragma unroll
  for (int j = 0; j < 16; ++j) {
    r[j] = x[gbase + j * 256 + tid];
    ls += r[j];
  }
  sRed[tid] = ls;
  __syncthreads();
  for (int st = 128; st > 0; st >>= 1) {
    if (tid < st) sRed[tid] += sRed[tid + st];
    __syncthreads();
  }
  const float inv_s = 1.0f / sRed[0];
  __syncthreads();

  float ls2 = 0.f;
  #pragma unroll
  for (int j = 0; j < 16; ++j) { r[j] *= inv_s; ls2 += r[j]; }
  sRed[tid] = ls2;
  __syncthreads();
  for (int st = 128; st > 0; st >>= 1) {
    if (tid < st) sRed[tid] += sRed[tid + st];
    __syncthreads();
  }
  const float mu = sRed[0] * (1.0f / (float)HW);
  __syncthreads();

  #pragma unroll
  for (int j = 0; j < 16; ++j) {
    int e = j * 256 + tid, row = e >> 6, col = e & 63;
    sXc[row * XS + col] = r[j] - mu;
  }
  __syncthreads();

  { // per-row means: 4 threads per row
    int rr = tid >> 2, part = tid & 3;
    float ps = 0.f;
    #pragma unroll
    for (int c = 0; c < 16; ++c) ps += sXc[rr * XS + part * 16 + c];
    sRed[tid] = ps;
  }
  __syncthreads();
  if (tid < 64)
    sRowM[tid] = (sRed[tid*4] + sRed[tid*4+1] + sRed[tid*4+2] + sRed[tid*4+3]) * (1.0f / 64.0f);
  __syncthreads();

  #pragma unroll
  for (int j = 0; j < 16; ++j) {
    int e = j * 256 + tid, row = e >> 6, col = e & 63;
    sTn[row * XS + col] = sXc[row * XS + col] - sRowM[row];
    sM[row * MS + 64 + col] = (row == col) ? 1.0f : 0.0f;  // identity (right half)
  }
  __syncthreads();

  // ---------------- Phase B: cov = Tn @ Tn^T / 63 + 1e-7*eps  (WMMA f32) ----------------
  // 16 output tiles of 16x16; 8 waves -> 2 tiles each. B fragment == A fragment of row-block j
  // since cov[h][g] = sum_w Tn[h][w]*Tn[g][w].
  #pragma unroll
  for (int t = 0; t < 2; ++t) {
    const int tile = wave + t * 8;
    const int ti = tile >> 2, tj = tile & 3;
    v8f acc = {};
    const int arow = (ti * 16 + l16) * XS + 2 * lhi;
    const int brow = (tj * 16 + l16) * XS + 2 * lhi;
    #pragma unroll
    for (int kb = 0; kb < 64; kb += 4) {
      v2f a = *(const v2f*)&sTn[arow + kb];
      v2f b = *(const v2f*)&sTn[brow + kb];
      acc = __builtin_amdgcn_wmma_f32_16x16x4_f32(false, a, false, b,
                                                  (short)0, acc, false, false);
    }
    #pragma unroll
    for (int q = 0; q < 8; ++q) {
      int row = ti * 16 + q + 8 * lhi;
      int col = tj * 16 + l16;
      float e = eps[gbase + row * 64 + col];
      sM[row * MS + col] = acc[q] * (1.0f / 63.0f) + e * 1e-7f;
    }
  }

  // ---------------- Phase C: Gauss-Jordan inversion of sM = [cov | I] ----------------
  for (int k = 0; k < NN; ++k) {
    __syncthreads();
    if (tid == 0) sPr = 1.0f / sM[k * MS + k];
    __syncthreads();
    if (tid < 128)       sM[k * MS + tid] *= sPr;          // scale pivot row (128 cols)
    else if (tid < 192)  sRowM[tid - 128] = sM[(tid - 128) * MS + k]; // cache pivot column
    __syncthreads();
    #pragma unroll 4
    for (int jj = 0; jj < 32; ++jj) {
      int e = jj * 256 + tid;
      int rr = e >> 7, c = e & 127;
      if (rr != k) sM[rr * MS + c] -= sRowM[rr] * sM[k * MS + c];
    }
  }
  __syncthreads();

  // ---------------- Phase D1: T1 = Xc @ Inv  (Inv = right half of sM) ----------------
  #pragma unroll
  for (int t = 0; t < 2; ++t) {
    const int tile = wave + t * 8;
    const int ti = tile >> 2, tj = tile & 3;
    v8f acc = {};
    const int arow = (ti * 16 + l16) * XS + 2 * lhi;
    #pragma unroll
    for (int kb = 0; kb < 64; kb += 4) {
      v2f a = *(const v2f*)&sXc[arow + kb];
      int k0 = kb + 2 * lhi;
      v2f b;
      b.x = sM[(k0    ) * MS + 64 + tj * 16 + l16];
      b.y = sM[(k0 + 1) * MS + 64 + tj * 16 + l16];
      acc = __builtin_amdgcn_wmma_f32_16x16x4_f32(false, a, false, b,
                                                  (short)0, acc, false, false);
    }
    #pragma unroll
    for (int q = 0; q < 8; ++q) {
      int row = ti * 16 + q + 8 * lhi;
      int col = tj * 16 + l16;
      sTn[row * XS + col] = acc[q];   // Tn is dead; reuse as T1
    }
  }
  __syncthreads();

  // ---------------- Phase D2: out = 1 / (T1 @ Xc + 1) ----------------
  #pragma unroll
  for (int t = 0; t < 2; ++t) {
    const int tile = wave + t * 8;
    const int ti = tile >> 2, tj = tile & 3;
    v8f acc = {};
    const int arow = (ti * 16 + l16) * XS + 2 * lhi;
    #pragma unroll
    for (int kb = 0; kb < 64; kb += 4) {
      v2f a = *(const v2f*)&sTn[arow + kb];
      int k0 = kb + 2 * lhi;
      v2f b;
      b.x = sXc[(k0    ) * XS + tj * 16 + l16];
      b.y = sXc[(k0 + 1) * XS + tj * 16 + l16];
      acc = __builtin_amdgcn_wmma_f32_16x16x4_f32(false, a, false, b,
                                                  (short)0, acc, false, false);
    }
    #pragma unroll
    for (int q = 0; q < 8; ++q) {
      int row = ti * 16 + q + 8 * lhi;
      int col = tj * 16 + l16;
      out[gbase + row * 64 + col] = 1.0f / (acc[q] + 1.0f);
    }
  }
}

extern "C" void kernel_launch(void* const* d_in, const int* in_sizes, int n_in,
                              void* d_out, int out_size, void* d_ws, size_t ws_size,
                              hipStream_t stream) {
  (void)in_sizes; (void)n_in; (void)out_size; (void)d_ws; (void)ws_size;
  const float* x   = (const float*)d_in[0];
  const float* eps = (const float*)d_in[1];
  float* out = (float*)d_out;
  dim3 grid(32 * 256);   // one workgroup per (b,c)
  dim3 block(256);       // 8 wave32 waves
  hipLaunchKernelGGL(gaussian_fit_kernel, grid, block, 0, stream, x, eps, out);
}